// Model_30605936951776
// MI455X (gfx1250) — compile-verified
//
#include <hip/hip_runtime.h>
#include <math.h>

#define S      160
#define NPIX   (S * S)        // 25600
#define NFACES 4096
#define NTILES (NPIX / 16)    // 1600 pixel tiles (M dimension)
#define FTILES (NFACES / 16)  // 256 face tiles  (N dimension)

typedef float f32x2 __attribute__((ext_vector_type(2)));
typedef float f32x8 __attribute__((ext_vector_type(8)));

// ---------------------------------------------------------------------------
// Kernel 1: project vertices.  ROT is rotation by pi/2 about Y (computed in
// float64 by the reference: diagonal term 1-(1-cos) = 1.1102230246251565e-16).
// vc = ROT@v + [0,0,2.732]; pix = (vc.xy/(vc.z*tan30) + 1) * 80
// ---------------------------------------------------------------------------
__global__ void k_project(const float* __restrict__ v, float* __restrict__ v2, int nv) {
  int i = blockIdx.x * blockDim.x + threadIdx.x;
  if (i >= nv) return;
  const float c2 = 1.1102230246251565e-16f;  // ROT diagonal
  const float VT = 0.57735026918962573f;     // tan(30 deg)
  float x = v[3*i], y = v[3*i+1], z = v[3*i+2];
  float vcx = c2 * x + z;
  float vcy = y;
  float vcz = -x + c2 * z + 2.732f;
  float inv = 1.0f / (vcz * VT);
  v2[2*i]   = (vcx * inv + 1.0f) * (S * 0.5f);
  v2[2*i+1] = (vcy * inv + 1.0f) * (S * 0.5f);
}

// ---------------------------------------------------------------------------
// Kernel 2: per-face edge planes.  Signed distance of edge (p0->p1) at pixel p:
//   dist = sg * (d.x*(py-p0.y) - d.y*(px-p0.x)) / (|d|+eps)
//        = A*px + B*py + C   with sg and 1/(|d|+eps) folded in.
// Invalid faces (|area|<=1e-6, incl. zero-padded faces) get A=B=0, C=-1e30
// so 1/(1+exp(dmin)) == 1 exactly -> face contributes nothing.
// planes layout: planes[(e*3 + k)*NFACES + face], e=edge 0..2, k=0(A),1(B),2(C)
// ---------------------------------------------------------------------------
__global__ void k_planes(const int* __restrict__ faces, const float* __restrict__ v2,
                         float* __restrict__ planes) {
  int f = blockIdx.x * blockDim.x + threadIdx.x;
  if (f >= NFACES) return;
  int ia = faces[3*f], ib = faces[3*f+1], ic = faces[3*f+2];
  float ax = v2[2*ia], ay = v2[2*ia+1];
  float bx = v2[2*ib], by = v2[2*ib+1];
  float cx = v2[2*ic], cy = v2[2*ic+1];
  float area = (bx-ax)*(cy-ay) - (by-ay)*(cx-ax);
  float sg   = (area >= 0.0f) ? 1.0f : -1.0f;
  bool valid = fabsf(area) > 1e-6f;
  float p0x[3] = {ax, bx, cx}, p0y[3] = {ay, by, cy};
  float p1x[3] = {bx, cx, ax}, p1y[3] = {by, cy, ay};
  #pragma unroll
  for (int e = 0; e < 3; ++e) {
    float dx = p1x[e]-p0x[e], dy = p1y[e]-p0y[e];
    float invl = sg / (sqrtf(dx*dx + dy*dy) + 1e-8f);
    float A = valid ? (-dy * invl)                     : 0.0f;
    float B = valid ? ( dx * invl)                     : 0.0f;
    float C = valid ? ((dy*p0x[e] - dx*p0y[e]) * invl) : -1e30f;
    planes[(e*3 + 0)*NFACES + f] = A;
    planes[(e*3 + 1)*NFACES + f] = B;
    planes[(e*3 + 2)*NFACES + f] = C;
  }
}

// ---------------------------------------------------------------------------
// B-matrix (4x16) load for one edge + WMMA.  Per ISA 32-bit layouts:
//   lanes 0-15:  VGPR0 = B[K=0][N]=coefA,  VGPR1 = B[K=1][N]=coefB
//   lanes 16-31: VGPR0 = B[K=2][N]=coefC,  VGPR1 = B[K=3][N]=0
// ---------------------------------------------------------------------------
__device__ __forceinline__
f32x8 edge_wmma(const float* __restrict__ planes, int e, int fidx, bool lo, f32x2 a) {
  int k0 = lo ? 0 : 2;
  f32x2 b;
  b.x = planes[(e*3 + k0)*NFACES + fidx];
  float b1 = planes[(e*3 + 1)*NFACES + fidx];
  b.y = lo ? b1 : 0.0f;
  f32x8 c = {0.f,0.f,0.f,0.f,0.f,0.f,0.f,0.f};
  // D = A(16x4) x B(4x16) + 0  -> v_wmma_f32_16x16x4_f32
  return __builtin_amdgcn_wmma_f32_16x16x4_f32(false, a, false, b, (short)0, c,
                                               false, false);
}

// ---------------------------------------------------------------------------
// Kernel 3: raster.  One wave per 16-pixel tile; loop 256 face tiles; per tile
// three v_wmma_f32_16x16x4_f32 (one per edge), elementwise min -> sigmoid ->
// product accumulation.  Epilogue uses 1 - sigmoid(d) == 1/(1 + exp(d)) with a
// raw v_rcp_f32 so the VALU cost per element is min3/exp/add/rcp/mul only.
// Butterfly product across each 16-lane half collapses the N (face) dimension.
// ---------------------------------------------------------------------------
__global__ void __launch_bounds__(256)
k_raster(const float* __restrict__ planes, const float* __restrict__ ref,
         float* __restrict__ out, float* __restrict__ tileLoss) {
  int lane = threadIdx.x & 31;
  int tile = blockIdx.x * (blockDim.x >> 5) + (threadIdx.x >> 5);
  bool lo = lane < 16;
  int  m  = lane & 15;

  // A matrix (16x4 f32): lanes 0-15 -> [px,py] (K=0,1); lanes 16-31 -> [1,0]
  int   pix0 = tile * 16 + m;
  float px = (float)(pix0 % S) + 0.5f;
  float py = (float)(pix0 / S) + 0.5f;
  f32x2 a;
  a.x = lo ? px : 1.0f;
  a.y = lo ? py : 0.0f;

  f32x8 acc;
  #pragma unroll
  for (int r = 0; r < 8; ++r) acc[r] = 1.0f;

  for (int t = 0; t < FTILES; ++t) {
    int fidx = t * 16 + m;
    f32x8 d0 = edge_wmma(planes, 0, fidx, lo, a);
    f32x8 d1 = edge_wmma(planes, 1, fidx, lo, a);
    f32x8 d2 = edge_wmma(planes, 2, fidx, lo, a);
    #pragma unroll
    for (int r = 0; r < 8; ++r) {
      float dm = fminf(fminf(d0[r], d1[r]), d2[r]);
      // 1 - sigmoid(dm) == 1/(1 + exp(dm)); v_exp_f32 + v_rcp_f32 fast path
      acc[r] *= __builtin_amdgcn_rcpf(1.0f + __expf(dm));
    }
  }

  // product over the 16 faces of each tile half (lanes 0-15: row r; 16-31: r+8)
  #pragma unroll
  for (int off = 1; off < 16; off <<= 1) {
    #pragma unroll
    for (int r = 0; r < 8; ++r)
      acc[r] *= __shfl_xor(acc[r], off, 32);
  }

  float lsum = 0.0f;
  if (m == 0) {
    int base = tile * 16 + (lo ? 0 : 8);
    #pragma unroll
    for (int r = 0; r < 8; ++r) {
      float img = 1.0f - acc[r];
      out[base + r] = img;
      float dd = img - ref[base + r];
      lsum += dd * dd;
    }
  }
  lsum += __shfl_xor(lsum, 16, 32);      // combine the two half-wave partials
  if (lane == 0) tileLoss[tile] = lsum;
}

// ---------------------------------------------------------------------------
// Kernel 4: deterministic loss reduction (single wave, fixed order).
// ---------------------------------------------------------------------------
__global__ void k_loss(const float* __restrict__ tileLoss, float* __restrict__ out) {
  int lane = threadIdx.x & 31;
  float s = 0.0f;
  for (int i = lane; i < NTILES; i += 32) s += tileLoss[i];
  #pragma unroll
  for (int off = 16; off > 0; off >>= 1) s += __shfl_xor(s, off, 32);
  if (lane == 0) out[NPIX] = s;
}

// ---------------------------------------------------------------------------
extern "C" void kernel_launch(void* const* d_in, const int* in_sizes, int n_in,
                              void* d_out, int out_size, void* d_ws, size_t ws_size,
                              hipStream_t stream) {
  const float* verts = (const float*)d_in[0];   // (1, nv, 3) f32
  const int*   faces = (const int*)d_in[1];     // (1, 4096, 3) i32
  const float* ref   = (const float*)d_in[2];   // (1, 160, 160) f32
  float* out = (float*)d_out;                   // 25600 image + 1 loss
  float* ws  = (float*)d_ws;

  float* v2       = ws;                          // 2*nv floats (<4096)
  float* planes   = ws + 4096;                   // 9*4096 floats
  float* tileLoss = ws + 4096 + 9 * NFACES;      // 1600 floats

  int nv = in_sizes[0] / 3;
  k_project<<<(nv + 255) / 256, 256, 0, stream>>>(verts, v2, nv);
  k_planes<<<(NFACES + 255) / 256, 256, 0, stream>>>(faces, v2, planes);
  k_raster<<<NTILES / 8, 256, 0, stream>>>(planes, ref, out, tileLoss);
  k_loss<<<1, 32, 0, stream>>>(tileLoss, out);
}